// HeteroGNN_7318624272988
// MI455X (gfx1250) — compile-verified
//
#include <hip/hip_runtime.h>

typedef float v2f __attribute__((ext_vector_type(2)));
typedef float v8f __attribute__((ext_vector_type(8)));

#define DD 128        // feature dim D == H
#define WSTRIDE 132   // padded LDS stride (banks: 132 mod 64 == 4 -> conflict-free frags)

__device__ __forceinline__ v8f wmma16x16x4(v2f a, v2f b, v8f c) {
  // D = A(16x4) * B(4x16) + C(16x16), full fp32 (matches reference precision)
  return __builtin_amdgcn_wmma_f32_16x16x4_f32(false, a, false, b, (short)0, c,
                                               false, false);
}

// ---------------------------------------------------------------------------
// Phase 1: edge scatter-add for segment mean. One wave32 per edge; each lane
// moves 4 consecutive floats (float4 gather, 4x global_atomic_add_f32).
// ---------------------------------------------------------------------------
__global__ __launch_bounds__(256) void scatter_mean_accum(
    const float* __restrict__ xsrc, const int* __restrict__ src,
    const int* __restrict__ dst, float* acc, float* cnt, int nedges) {
  int gid = blockIdx.x * 256 + threadIdx.x;
  int e = gid >> 5;
  if (e >= nedges) return;
  int lane = threadIdx.x & 31;
  int s = src[e];          // wave-uniform -> scalar load
  int d = dst[e];
  const float4* row = (const float4*)(xsrc + (size_t)s * DD);
  float4 v = row[lane];    // 128 floats / 32 lanes = 1 float4 per lane
  float* a = acc + (size_t)d * DD + lane * 4;
  unsafeAtomicAdd(a + 0, v.x);
  unsafeAtomicAdd(a + 1, v.y);
  unsafeAtomicAdd(a + 2, v.z);
  unsafeAtomicAdd(a + 3, v.w);
  if (lane == 0) unsafeAtomicAdd(cnt + d, 1.0f);
}

// Cooperative load of a 128x128 row-major weight into transposed padded LDS:
// ldsW[n*WSTRIDE + k] = W[k*DD + n]. Global reads coalesced.
__device__ __forceinline__ void stage_weight_T(float* ldsW,
                                               const float* __restrict__ W) {
  for (int i = threadIdx.x; i < DD * DD; i += 256) {
    int k = i >> 7;
    int n = i & (DD - 1);
    ldsW[n * WSTRIDE + k] = W[i];
  }
}

// ---------------------------------------------------------------------------
// Phase 2: h = (acc/cnt) @ Wl + bl + x @ Wr  per 16-row tile, one wave/tile.
// Writes h in place over acc (tile rows are wave-exclusive).
// Wl/Wr transposed+padded in ~132 KB dynamic LDS; B-frags are single
// conflict-free ds_load_b64 with compile-time offsets.
// ---------------------------------------------------------------------------
__global__ __launch_bounds__(256) void sage_h_kernel(
    const float* agg_in, const float* __restrict__ cnt,
    const float* __restrict__ x, const float* __restrict__ Wl,
    const float* __restrict__ bl, const float* __restrict__ Wr,
    float* h_out, int ntiles) {
  extern __shared__ float lds[];            // [2 * 128 * WSTRIDE]
  float* ldsWl = lds;
  float* ldsWr = lds + DD * WSTRIDE;
  stage_weight_T(ldsWl, Wl);
  stage_weight_T(ldsWr, Wr);
  __syncthreads();

  int wave = threadIdx.x >> 5;
  int lane = threadIdx.x & 31;
  int tile = blockIdx.x * 8 + wave;         // wave-uniform; EXEC stays all-ones
  if (tile >= ntiles) return;

  int row0 = tile * 16;
  int mrow = lane & 15;                     // A row / C column within tile
  int khalf = lane >> 4;                    // 0: K=0,1  1: K=2,3 (ISA 16x4 layout)
  int r = row0 + mrow;
  float invc = 1.0f / fmaxf(cnt[r], 1.0f);

  const v2f* arow = (const v2f*)(agg_in + (size_t)r * DD) + khalf;
  const v2f* xrow = (const v2f*)(x + (size_t)r * DD) + khalf;

  // Per-lane fragment base pointers: B frag (kk,nt) = bptr[nt][kk*2]
  const v2f* blp[8];
  const v2f* brp[8];
#pragma unroll
  for (int nt = 0; nt < 8; ++nt) {
    int ncol = nt * 16 + mrow;
    blp[nt] = (const v2f*)(ldsWl + ncol * WSTRIDE) + khalf;
    brp[nt] = (const v2f*)(ldsWr + ncol * WSTRIDE) + khalf;
  }

  v8f c[8];
#pragma unroll
  for (int nt = 0; nt < 8; ++nt) {
    float bv = bl[nt * 16 + mrow];          // bias depends on N only
#pragma unroll
    for (int j = 0; j < 8; ++j) c[nt][j] = bv;
  }

#pragma unroll 4
  for (int kk = 0; kk < 32; ++kk) {         // K = 128 in steps of 4
    v2f a1 = arow[kk * 2];                  // agg frag: k0 + 2*khalf + {0,1}
    a1 *= invc;                             // segment mean
    v2f a2 = xrow[kk * 2];                  // x frag
#pragma unroll
    for (int nt = 0; nt < 8; ++nt) {
      c[nt] = wmma16x16x4(a1, blp[nt][kk * 2], c[nt]);
      c[nt] = wmma16x16x4(a2, brp[nt][kk * 2], c[nt]);
    }
  }

#pragma unroll
  for (int nt = 0; nt < 8; ++nt) {
#pragma unroll
    for (int v = 0; v < 8; ++v) {
      int m = v + 8 * khalf;                // C layout: VGPR v -> row v / v+8
      h_out[(size_t)(row0 + m) * DD + nt * 16 + mrow] = c[nt][v];
    }
  }
}

// ---------------------------------------------------------------------------
// Phase 3: out = relu(h @ W + b), one wave per 16-row tile, W^T in ~66 KB LDS.
// ---------------------------------------------------------------------------
__global__ __launch_bounds__(256) void sage_out_kernel(
    const float* __restrict__ h, const float* __restrict__ W,
    const float* __restrict__ b, float* __restrict__ out, int ntiles) {
  extern __shared__ float lds[];            // [128 * WSTRIDE]
  stage_weight_T(lds, W);
  __syncthreads();

  int wave = threadIdx.x >> 5;
  int lane = threadIdx.x & 31;
  int tile = blockIdx.x * 8 + wave;
  if (tile >= ntiles) return;

  int row0 = tile * 16;
  int mrow = lane & 15;
  int khalf = lane >> 4;
  int r = row0 + mrow;
  const v2f* hrow = (const v2f*)(h + (size_t)r * DD) + khalf;

  const v2f* bp[8];
#pragma unroll
  for (int nt = 0; nt < 8; ++nt) {
    int ncol = nt * 16 + mrow;
    bp[nt] = (const v2f*)(lds + ncol * WSTRIDE) + khalf;
  }

  v8f c[8];
#pragma unroll
  for (int nt = 0; nt < 8; ++nt) {
    float bv = b[nt * 16 + mrow];
#pragma unroll
    for (int j = 0; j < 8; ++j) c[nt][j] = bv;
  }

#pragma unroll 4
  for (int kk = 0; kk < 32; ++kk) {
    v2f a = hrow[kk * 2];
#pragma unroll
    for (int nt = 0; nt < 8; ++nt) {
      c[nt] = wmma16x16x4(a, bp[nt][kk * 2], c[nt]);
    }
  }

#pragma unroll
  for (int nt = 0; nt < 8; ++nt) {
#pragma unroll
    for (int v = 0; v < 8; ++v) {
      int m = v + 8 * khalf;
      out[(size_t)(row0 + m) * DD + nt * 16 + mrow] = fmaxf(c[nt][v], 0.0f);
    }
  }
}

// ---------------------------------------------------------------------------
extern "C" void kernel_launch(void* const* d_in, const int* in_sizes, int n_in,
                              void* d_out, int out_size, void* d_ws,
                              size_t ws_size, hipStream_t stream) {
  const float* x_user = (const float*)d_in[0];
  const float* x_item = (const float*)d_in[1];
  const int* esrc_ui  = (const int*)d_in[2];
  const int* edst_ui  = (const int*)d_in[3];
  const int* esrc_iu  = (const int*)d_in[4];
  const int* edst_iu  = (const int*)d_in[5];
  const float* Wl_ui  = (const float*)d_in[6];
  const float* bl_ui  = (const float*)d_in[7];
  const float* Wr_ui  = (const float*)d_in[8];
  const float* Wl_iu  = (const float*)d_in[9];
  const float* bl_iu  = (const float*)d_in[10];
  const float* Wr_iu  = (const float*)d_in[11];
  const float* W_user = (const float*)d_in[12];
  const float* b_user = (const float*)d_in[13];
  const float* W_item = (const float*)d_in[14];
  const float* b_item = (const float*)d_in[15];

  const int NU = in_sizes[0] / DD;     // 200000
  const int NI = in_sizes[1] / DD;     // 100000
  const int E  = in_sizes[2];          // 600000

  float* ws = (float*)d_ws;
  float* acc_user = ws;                                // [NU*128] -> later h_user
  float* acc_item = acc_user + (size_t)NU * DD;        // [NI*128] -> later h_item
  float* cnt_user = acc_item + (size_t)NI * DD;        // [NU]
  float* cnt_item = cnt_user + NU;                     // [NI]
  size_t zero_bytes = ((size_t)(NU + NI) * DD + NU + NI) * sizeof(float);
  hipMemsetAsync(d_ws, 0, zero_bytes, stream);

  // Segment-sum scatter (one wave per edge)
  int eblocks = (E * 32 + 255) / 256;
  scatter_mean_accum<<<eblocks, 256, 0, stream>>>(x_user, esrc_ui, edst_ui,
                                                  acc_item, cnt_item, E);
  scatter_mean_accum<<<eblocks, 256, 0, stream>>>(x_item, esrc_iu, edst_iu,
                                                  acc_user, cnt_user, E);

  int tiles_u = NU / 16, tiles_i = NI / 16;
  size_t lds2 = (size_t)2 * DD * WSTRIDE * sizeof(float);   // ~132 KB
  sage_h_kernel<<<(tiles_i + 7) / 8, 256, lds2, stream>>>(
      acc_item, cnt_item, x_item, Wl_ui, bl_ui, Wr_ui, acc_item, tiles_i);
  sage_h_kernel<<<(tiles_u + 7) / 8, 256, lds2, stream>>>(
      acc_user, cnt_user, x_user, Wl_iu, bl_iu, Wr_iu, acc_user, tiles_u);

  float* out = (float*)d_out;
  size_t lds1 = (size_t)DD * WSTRIDE * sizeof(float);       // ~66 KB
  sage_out_kernel<<<(tiles_u + 7) / 8, 256, lds1, stream>>>(
      acc_user, W_user, b_user, out, tiles_u);
  sage_out_kernel<<<(tiles_i + 7) / 8, 256, lds1, stream>>>(
      acc_item, W_item, b_item, out + (size_t)NU * DD, tiles_i);
}